// SSL_6459630813843
// MI455X (gfx1250) — compile-verified
//
#include <hip/hip_runtime.h>
#include <math.h>

// ---------------------------------------------------------------------------
// SSL stroke-label loss on MI455X (gfx1250, wave32).
//   score[m,n] = ||p_n||^2 - 2 t_m . p_n  ==  A x B with K=4 via
//   V_WMMA_F32_16X16X4_F32  (A row = [-2tx,-2ty,1,0], B col = [px,py,|p|^2,0])
//   argmin_n score == 1-NN; then scatter labels, then deterministic BCE sum.
// Round-2 changes: 4x unrolled tile loop (4 loads + 4 wmma in flight per
// iteration) and branch-free u64-packed argmin epilogue.
// ---------------------------------------------------------------------------

typedef __attribute__((ext_vector_type(2))) float v2f;
typedef __attribute__((ext_vector_type(8))) float v8f;

#define BATCH 64
#define NPRED 2048
#define MTARG 512

__global__ void ssl_zero_labels(float* __restrict__ labels) {
    int i = blockIdx.x * blockDim.x + threadIdx.x;
    if (i < BATCH * NPRED) labels[i] = 0.0f;
}

// One wave = one 16-row strip of targs for one batch, sweeping all 128 n-tiles.
// D-tile layout (f32 16x16): VGPR r -> rows (r, r+8); lanes 0-15 / 16-31 are
// the 16 tile columns for each row group. Running (min,idx) kept per VGPR slot.
__global__ __launch_bounds__(32)
void ssl_nn_scatter(const float* __restrict__ preds,
                    const float* __restrict__ targs,
                    float* __restrict__ labels) {
    const int wave  = blockIdx.x;        // 0 .. BATCH*(MTARG/16)-1
    const int b     = wave >> 5;         // batch index (MTARG/16 == 32)
    const int strip = wave & 31;         // which 16-row strip of targs
    const int lane  = threadIdx.x;       // 0..31, wave32
    const int hi    = lane >> 4;         // half of the wave
    const int col   = lane & 15;

    const float* __restrict__ pb = preds + (size_t)b * NPRED * 3;

    // --- A operand: 16x4 f32.  Lanes 0-15 hold K=0..1 (VGPR0/1), lanes 16-31
    //     hold K=2..3.  A[m] = [-2tx, -2ty, 1, 0].
    const int mrow = strip * 16 + col;
    const float tx = targs[(b * MTARG + mrow) * 3 + 0];
    const float ty = targs[(b * MTARG + mrow) * 3 + 1];
    v2f a;
    a.x = hi ? 1.0f : -2.0f * tx;   // K=0 (lo half) / K=2 (hi half)
    a.y = hi ? 0.0f : -2.0f * ty;   // K=1 (lo half) / K=3 (hi half)

    float vmin[8];
    int   vidx[8];
#pragma unroll
    for (int r = 0; r < 8; ++r) { vmin[r] = 3.402823466e38f; vidx[r] = 0; }

    // 4x unroll: issue 4 independent global loads, then 4 wmma + min-updates.
    for (int t = 0; t < NPRED / 16; t += 4) {
        float px[4], py[4];
#pragma unroll
        for (int j = 0; j < 4; ++j) {
            const int n = (t + j) * 16 + col;
            px[j] = pb[n * 3 + 0];
            py[j] = pb[n * 3 + 1];
        }
#pragma unroll
        for (int j = 0; j < 4; ++j) {
            // --- B operand: 4x16 f32. Lanes 0-15 hold K=0..1 rows, lanes
            //     16-31 hold K=2..3 rows; column = lane&15.
            //     B[:,n] = [px, py, |p|^2, 0].
            v2f bb;
            bb.x = hi ? (px[j] * px[j] + py[j] * py[j]) : px[j];
            bb.y = hi ? 0.0f                            : py[j];
            v8f c = {0.0f, 0.0f, 0.0f, 0.0f, 0.0f, 0.0f, 0.0f, 0.0f};
            v8f d = __builtin_amdgcn_wmma_f32_16x16x4_f32(
                /*neg_a=*/false, a, /*neg_b=*/false, bb,
                /*c_mod=*/(short)0, c, /*reuse_a=*/false, /*reuse_b=*/false);
            const int nidx = (t + j) * 16 + col;
#pragma unroll
            for (int r = 0; r < 8; ++r) {
                const float dv = d[r];
                if (dv < vmin[r]) { vmin[r] = dv; vidx[r] = nidx; }
            }
        }
    }

    // --- Branch-free reduction across the 16 lanes of each half.
    //     Pack (sortable f32 key || index) into u64; unsigned min implements
    //     "smaller score, then smaller index" == argmin first-occurrence.
#pragma unroll
    for (int r = 0; r < 8; ++r) {
        const unsigned u   = __float_as_uint(vmin[r]);
        const unsigned key = (u & 0x80000000u) ? ~u : (u | 0x80000000u);
        unsigned long long pk =
            ((unsigned long long)key << 32) | (unsigned)vidx[r];
#pragma unroll
        for (int off = 8; off > 0; off >>= 1) {
            const unsigned long long o =
                (unsigned long long)__shfl_xor((long long)pk, off, 16);
            pk = (o < pk) ? o : pk;
        }
        // lanes 0-15 carry row strip*16+r, lanes 16-31 carry strip*16+r+8
        if (col == r) {
            const int m = strip * 16 + r + hi * 8;
            if (targs[(b * MTARG + m) * 3 + 2] != 0.0f) {
                const int nn = (int)(pk & 0xFFFFFFFFull);
                labels[b * NPRED + nn] = 1.0f;   // benign race: same value
            }
        }
    }
}

// BCE per element -> one partial per 256-thread block (fixed-order tree).
__global__ void ssl_bce_partial(const float* __restrict__ preds,
                                const float* __restrict__ labels,
                                float* __restrict__ partials, int total) {
    __shared__ float sdata[256];
    const int i = blockIdx.x * blockDim.x + threadIdx.x;
    float v = 0.0f;
    if (i < total) {
        float p = preds[i * 3 + 2];
        p = fminf(fmaxf(p, 1e-7f), 1.0f - 1e-7f);
        const float l = labels[i];
        v = (l > 0.5f) ? -logf(p) : -log1pf(-p);
    }
    sdata[threadIdx.x] = v;
    __syncthreads();
#pragma unroll
    for (int s = 128; s > 0; s >>= 1) {
        if (threadIdx.x < s) sdata[threadIdx.x] += sdata[threadIdx.x + s];
        __syncthreads();
    }
    if (threadIdx.x == 0) partials[blockIdx.x] = sdata[0];
}

// Single-block deterministic final sum; scale by 1/N (mean over N, sum over B).
__global__ void ssl_final_reduce(const float* __restrict__ partials,
                                 float* __restrict__ out,
                                 int n_partials, float inv_n) {
    __shared__ float sdata[256];
    float v = 0.0f;
    for (int i = threadIdx.x; i < n_partials; i += 256) v += partials[i];
    sdata[threadIdx.x] = v;
    __syncthreads();
#pragma unroll
    for (int s = 128; s > 0; s >>= 1) {
        if (threadIdx.x < s) sdata[threadIdx.x] += sdata[threadIdx.x + s];
        __syncthreads();
    }
    if (threadIdx.x == 0) out[0] = sdata[0] * inv_n;
}

extern "C" void kernel_launch(void* const* d_in, const int* in_sizes, int n_in,
                              void* d_out, int out_size, void* d_ws, size_t ws_size,
                              hipStream_t stream) {
    const float* preds = (const float*)d_in[0];   // [B, N, 3]
    const float* targs = (const float*)d_in[1];   // [B, M, 3]
    float* out = (float*)d_out;                   // [1]

    float* labels   = (float*)d_ws;               // B*N floats (512 KB)
    float* partials = labels + BATCH * NPRED;     // 512 floats

    const int total    = BATCH * NPRED;           // 131072
    const int n_blocks = total / 256;             // 512

    ssl_zero_labels<<<(total + 255) / 256, 256, 0, stream>>>(labels);
    ssl_nn_scatter<<<BATCH * (MTARG / 16), 32, 0, stream>>>(preds, targs, labels);
    ssl_bce_partial<<<n_blocks, 256, 0, stream>>>(preds, labels, partials, total);
    ssl_final_reduce<<<1, 256, 0, stream>>>(partials, out, n_blocks,
                                            1.0f / (float)NPRED);
}